// NaiveBayes_90134183674440
// MI455X (gfx1250) — compile-verified
//
#include <hip/hip_runtime.h>
#include <stdint.h>

// Problem geometry (fixed by the reference).
#define BATCH   16
#define TLEN    8192                 // output positions per row; tokens per row = TLEN-1
#define ALPHA   256
#define CHUNKS  64
#define CS      (TLEN / CHUNKS)      // 128 positions per chunk

typedef unsigned int u32x4 __attribute__((ext_vector_type(4)));
typedef unsigned int u32x8 __attribute__((ext_vector_type(8)));

// ---------------------------------------------------------------------------
// Pass 1: per-chunk token histograms -> ws[(b*CHUNKS + c)*ALPHA + a]
// Tokens staged global->LDS with the ASYNCcnt async-copy path, then ds_add
// atomics build the 256-bin histogram.
// ---------------------------------------------------------------------------
__global__ void nb_chunk_hist(const int* __restrict__ x, unsigned* __restrict__ ws) {
    __shared__ unsigned h[ALPHA];
    __shared__ int      t_tok[CS];
    const int c = blockIdx.x, b = blockIdx.y, tid = threadIdx.x;

    if (tid < CS) {
        const int j = c * CS + tid;                 // global token index in [0, TLEN-1)
        if (j < TLEN - 1) {
            const unsigned lds = (unsigned)(uintptr_t)&t_tok[tid];
            const unsigned long long g =
                (unsigned long long)(uintptr_t)(x + (size_t)b * TLEN + 1 + j);
            asm volatile("global_load_async_to_lds_b32 %0, %1, off"
                         :: "v"(lds), "v"(g) : "memory");
        } else {
            t_tok[tid] = -1;                        // tail sentinel (never counted)
        }
    }
    h[tid] = 0u;
    asm volatile("s_wait_asynccnt 0" ::: "memory");
    __syncthreads();

    if (tid < CS) {
        const int tok = t_tok[tid];
        if (tok >= 0) atomicAdd(&h[tok & (ALPHA - 1)], 1u);
    }
    __syncthreads();
    ws[(unsigned)(b * CHUNKS + c) * ALPHA + tid] = h[tid];
}

// ---------------------------------------------------------------------------
// Pass 2: per-row, per-symbol exclusive scan across chunks (start counts).
// ---------------------------------------------------------------------------
__global__ void nb_chunk_scan(unsigned* __restrict__ ws) {
    const int b = blockIdx.x, a = threadIdx.x;
    unsigned run = 0u;
    for (int c = 0; c < CHUNKS; ++c) {
        const unsigned idx = (unsigned)(b * CHUNKS + c) * ALPHA + a;
        const unsigned v = ws[idx];
        ws[idx] = run;          // exclusive prefix = counts before this chunk
        run += v;
    }
}

// ---------------------------------------------------------------------------
// Pass 3: main streaming kernel. Thread a owns symbol a's count in a register.
// The chunk's 128 tokens are DMA'd into LDS by the Tensor Data Mover
// (tensor_load_to_lds, TENSORcnt), issued once by wave 0. OOB reads past
// tensor_dim0 zero-fill, which covers the 127-token tail of the last chunk.
// out[b][t][a] = log(base[a]+cnt) - log(sum(base)+t); at most one lane per
// position recomputes its log.
// ---------------------------------------------------------------------------
__global__ void nb_main(const int* __restrict__ x, const float* __restrict__ base,
                        const unsigned* __restrict__ ws, float* __restrict__ out) {
    __shared__ int   s_tok[CS];
    __shared__ float s_denom[CS];
    __shared__ float s_red[ALPHA];
    const int c = blockIdx.x, b = blockIdx.y, a = threadIdx.x;
    const int lastValid = (TLEN - 1) - c * CS;   // tokens consumable in this chunk

    // ---- TDM: descriptor-driven 1-D tile load, global -> LDS (wave 0 only) ----
    if (a < 32) {
        const unsigned lds_addr = (unsigned)(uintptr_t)s_tok;   // low 32 bits = LDS offset
        const unsigned long long gaddr =
            (unsigned long long)(uintptr_t)(x + (size_t)b * TLEN + 1 + (size_t)c * CS);
        const unsigned tdim0 = (unsigned)((lastValid < CS) ? lastValid : CS); // OOB->0 fill
        const unsigned tdim1 = 1u;
        const unsigned tile0 = (unsigned)CS;

        u32x4 g0;
        g0.x = 1u;                                   // count=1, user descriptor
        g0.y = lds_addr;                             // lds_addr[31:0]
        g0.z = (unsigned)(gaddr & 0xFFFFFFFFull);    // global_addr[31:0]
        g0.w = (unsigned)((gaddr >> 32) & 0x01FFFFFFull) | (2u << 30); // addr[56:32], type=2

        u32x8 g1;
        g1.s0 = (2u << 16);                          // data_size=2 -> 4-byte elements
        g1.s1 = (tdim0 & 0xFFFFu) << 16;             // tensor_dim0[15:0]
        g1.s2 = ((tdim0 >> 16) & 0xFFFFu) | ((tdim1 & 0xFFFFu) << 16);
        g1.s3 = ((tdim1 >> 16) & 0xFFFFu) | ((tile0 & 0xFFFFu) << 16); // tile_dim0
        g1.s4 = 0u;                                  // tile_dim1=0, tile_dim2=0 (unused)
        g1.s5 = tile0;                               // tensor_dim0_stride[31:0] (benign)
        g1.s6 = 0u;
        g1.s7 = 0u;

        u32x4 g2 = (u32x4)(0u);
        u32x4 g3 = (u32x4)(0u);

        asm volatile("tensor_load_to_lds %0, %1, %2, %3"
                     :: "s"(g0), "s"(g1), "s"(g2), "s"(g3) : "memory");
    }

    // ---- sum(base) block reduction (denominator) ----
    s_red[a] = base[a];
    __syncthreads();
    for (int off = ALPHA / 2; off > 0; off >>= 1) {
        if (a < off) s_red[a] += s_red[a + off];
        __syncthreads();
    }
    const float sumBase = s_red[0];

    // ---- per-position denominator log table: log(sum(base) + t) ----
    if (a < CS) s_denom[a] = __logf(sumBase + (float)(c * CS + a));

    // Wait for the TDM transfer (issuing wave), then publish LDS to all waves.
    if (a < 32) asm volatile("s_wait_tensorcnt 0x0" ::: "memory");
    __syncthreads();

    // ---- streaming loop: 128 positions, 1 KiB coalesced store per step ----
    unsigned cnt = ws[(unsigned)(b * CHUNKS + c) * ALPHA + a];
    const float baseA = base[a];
    float myLog = __logf(baseA + (float)cnt);
    size_t o = ((size_t)b * TLEN + (size_t)c * CS) * ALPHA + (size_t)a;

    #pragma unroll 4
    for (int i = 0; i < CS; ++i, o += ALPHA) {
        out[o] = myLog - s_denom[i];             // coalesced b32 store
        const int tok = s_tok[i];                // uniform LDS broadcast read
        if (i < lastValid && a == tok) {         // at most one lane per block
            cnt += 1u;
            myLog = __logf(baseA + (float)cnt);
        }
    }
}

// ---------------------------------------------------------------------------
extern "C" void kernel_launch(void* const* d_in, const int* in_sizes, int n_in,
                              void* d_out, int out_size, void* d_ws, size_t ws_size,
                              hipStream_t stream) {
    (void)in_sizes; (void)n_in; (void)out_size; (void)ws_size;
    const int*   x    = (const int*)d_in[0];     // (16, 8192) int32
    const float* base = (const float*)d_in[1];   // (256,) float32
    float*       out  = (float*)d_out;           // (16, 8192, 256) float32
    unsigned*    ws   = (unsigned*)d_ws;         // 16*64*256 u32 = 1 MiB scratch

    dim3 grid(CHUNKS, BATCH);
    nb_chunk_hist<<<grid, ALPHA, 0, stream>>>(x, ws);
    nb_chunk_scan<<<BATCH, ALPHA, 0, stream>>>(ws);
    nb_main<<<grid, ALPHA, 0, stream>>>(x, base, ws, out);
}